// RegularizedCRFLoss_14654428414060
// MI455X (gfx1250) — compile-verified
//
#include <hip/hip_runtime.h>

typedef __attribute__((ext_vector_type(2))) float v2f;
typedef __attribute__((ext_vector_type(8))) float v8f;

#if defined(__AMDGCN__) && __has_builtin(__builtin_amdgcn_global_load_async_to_lds_b32)
#define USE_ASYNC_LDS 1
#else
#define USE_ASYNC_LDS 0
#endif

typedef __attribute__((address_space(1))) int* gptr_t;
typedef __attribute__((address_space(3))) int* lptr_t;

static constexpr int kN = 4;
static constexpr int kC = 21;
static constexpr int kH = 96;
static constexpr int kW = 96;
static constexpr int kR = 5;      // kernels_radius
static constexpr int kD = 11;     // 2r+1
static constexpr int kCP = 24;    // classes padded to WMMA K = 24 (21..23 = 0)
static constexpr int kCols = 48;  // 2 tiles * 16 + 2*5 halo = 42, padded to 48
static constexpr int kKS = 6;     // six chained v_wmma_f32_16x16x4_f32 -> K = 24

static constexpr int kYSlots = kD * kCols * kCP;  // 12672 floats (50.7 KB)
static constexpr int kRSlots = kD * kCols * 4;    // 2112 floats  ( 8.4 KB)

__global__ void __launch_bounds__(64)
crf_loss_wmma(const float* __restrict__ Y, const float* __restrict__ RGB,
              float* __restrict__ out)
{
  __shared__ __align__(16) float sY[kYSlots];  // [row][col][class]
  __shared__ __align__(16) float sR[kRSlots];  // [row][col][ch(4)]

  const int tid = threadIdx.x;
  int b = blockIdx.x;
  const int pair = b % 3;  b /= 3;   // 32-column pair within the row
  const int h    = b % kH; b /= kH;
  const int n    = b;
  const int x0   = pair * 32;        // first center column of this pair

  // ---------------- stage Y halo strip (zero-padded) into LDS ----------------
  for (int s = tid; s < kYSlots; s += 64) {
    const int c   = s % kCP;
    const int col = (s / kCP) % kCols;
    const int row = s / (kCP * kCols);
    const int hh  = h + row - kR;
    const int ww  = x0 - kR + col;
    const bool in = (c < kC) && ((unsigned)hh < (unsigned)kH) &&
                    ((unsigned)ww < (unsigned)kW);
#if USE_ASYNC_LDS
    if (in) {
      const float* gp = Y + (((n * kC + c) * kH + hh) * kW + ww);
      __builtin_amdgcn_global_load_async_to_lds_b32(
          (gptr_t)(void*)gp, (lptr_t)(void*)(sY + s), 0, 0);
    } else {
      sY[s] = 0.0f;
    }
#else
    sY[s] = in ? Y[((n * kC + c) * kH + hh) * kW + ww] : 0.0f;
#endif
  }

  // ---------------- stage RGB halo strip (zero-padded) into LDS ----------------
  for (int s = tid; s < kRSlots; s += 64) {
    const int ch  = s % 4;
    const int col = (s / 4) % kCols;
    const int row = s / (4 * kCols);
    const int hh  = h + row - kR;
    const int ww  = x0 - kR + col;
    const bool in = (ch < 3) && ((unsigned)hh < (unsigned)kH) &&
                    ((unsigned)ww < (unsigned)kW);
#if USE_ASYNC_LDS
    if (in) {
      const float* gp = RGB + (((n * 3 + ch) * kH + hh) * kW + ww);
      __builtin_amdgcn_global_load_async_to_lds_b32(
          (gptr_t)(void*)gp, (lptr_t)(void*)(sR + s), 0, 0);
    } else {
      sR[s] = 0.0f;
    }
#else
    sR[s] = in ? RGB[((n * 3 + ch) * kH + hh) * kW + ww] : 0.0f;
#endif
  }

#if USE_ASYNC_LDS
#if __has_builtin(__builtin_amdgcn_s_wait_asynccnt)
  __builtin_amdgcn_s_wait_asynccnt(0);
#else
  asm volatile("s_wait_asynccnt 0" ::: "memory");
#endif
#endif
  __syncthreads();

  // ---------------- compute: one wave per 16-pixel tile ----------------
  const int lane = tid & 31;
  const int wv   = tid >> 5;           // 0..1 : tile within the pair
  const int half = lane >> 4;          // 0: lanes 0-15, 1: lanes 16-31
  const int l15  = lane & 15;
  const int xl   = 16 * wv;            // tile's first center col (local, halo space +kR)
  const int x    = x0 + xl;            // tile's first center col (global)

  // A matrix: center strip, f32 16x16x4 layout
  // lanes 0-15: M=lane, {K=4kk+0,4kk+1}; lanes 16-31: M=lane-16, {K=4kk+2,4kk+3}
  v2f A[kKS];
  const float* aptr = &sY[((kR * kCols) + (kR + xl + l15)) * kCP];
#pragma unroll
  for (int kk = 0; kk < kKS; ++kk)
    A[kk] = *(const v2f*)(aptr + 4 * kk + 2 * half);   // ds_load_b64

  // center rgb (pre-scaled by 1/sigma_rgb = 10) for this lane's 8 M rows
  float rc0[8], rc1[8], rc2[8];
#pragma unroll
  for (int v = 0; v < 8; ++v) {
    const float* rp = &sR[((kR * kCols) + (kR + xl + v + 8 * half)) * 4];
    rc0[v] = rp[0] * 10.0f; rc1[v] = rp[1] * 10.0f; rc2[v] = rp[2] * 10.0f;
  }

  const float cy = (float)h * (1.0f / 6.0f);
  float acc_loss = 0.0f;  // accumulates  K * (1 - S)

  for (int dy = 0; dy < kD; ++dy) {
    const int hn = h + dy - kR;
    const bool row_in = (unsigned)hn < (unsigned)kH;
    const float uy = row_in ? (float)hn * (1.0f / 6.0f) : 0.0f;  // padded y-feature
    const bool center_row = (dy == kR);

#pragma unroll
    for (int t = 0; t < 2; ++t) {
      const int nglob = l15 + 16 * t;   // 0..31
      const int colL  = xl + nglob;     // LDS col (maps to global w = x - 5 + nglob)
      const int wn    = x - kR + nglob;

      v2f Bv[kKS];
      const float* bptr = &sY[((dy * kCols) + colL) * kCP];
#pragma unroll
      for (int kk = 0; kk < kKS; ++kk)
        Bv[kk] = *(const v2f*)(bptr + 4 * kk + 2 * half);  // ds_load_b64

      v8f acc = {0.f, 0.f, 0.f, 0.f, 0.f, 0.f, 0.f, 0.f};
#pragma unroll
      for (int kk = 0; kk < kKS; ++kk)
        acc = __builtin_amdgcn_wmma_f32_16x16x4_f32(
            false, A[kk], false, Bv[kk], (short)0, acc, false, false);

      const bool nb_in = row_in && ((unsigned)wn < (unsigned)kW);
      const float ux   = nb_in ? (float)wn * (1.0f / 6.0f) : 0.0f;  // padded x-feature
      const float uyv  = nb_in ? uy : 0.0f;
      const float* rp  = &sR[((dy * kCols) + colL) * 4];            // staged 0 if OOB
      const float rn0 = rp[0] * 10.0f, rn1 = rp[1] * 10.0f, rn2 = rp[2] * 10.0f;

      // Branch-free weighting of the 8 Gram entries this lane owns.
#pragma unroll
      for (int v = 0; v < 8; ++v) {
        const int m   = v + 8 * half;   // D layout: M = v + 8*half, N = l15
        const int dxn = nglob - m;      // dx + 5
        const float cx  = (float)(x + m) * (1.0f / 6.0f);
        const float ex  = ux - cx;      // = dx/6 in-bounds, = -wc/6 padded
        const float ey  = uyv - cy;     // = (dy-5)/6 in-bounds, = -h/6 padded
        const float d2xy = ex * ex + ey * ey;
        const float e0 = rn0 - rc0[v];
        const float e1 = rn1 - rc1[v];
        const float e2 = rn2 - rc2[v];
        const float d2rgb = e0 * e0 + e1 * e1 + e2 * e2;
        const float exy  = __expf(-0.5f * d2xy);
        const float ergb = __expf(-0.5f * d2rgb);
        // K = 0.9*exp(-(d2xy+d2rgb)/2) + 0.1*exp(-d2xy/2)
        float K = exy * (0.9f * ergb + 0.1f);
        const bool valid = ((unsigned)dxn <= 10u) && !(center_row && dxn == kR);
        K = valid ? K : 0.0f;           // off-band entries + center tap -> 0
        acc_loss += K * (1.0f - acc[v]);
      }
    }
  }

  // wave32 reduction, one atomic per wave
#pragma unroll
  for (int mask = 16; mask > 0; mask >>= 1)
    acc_loss += __shfl_xor(acc_loss, mask, 32);
  if (lane == 0)
    atomicAdd(out, acc_loss * (1.0f / (float)(kN * kH * kW)));
}

__global__ void zero_out_kernel(float* out) { out[0] = 0.0f; }

extern "C" void kernel_launch(void* const* d_in, const int* in_sizes, int n_in,
                              void* d_out, int out_size, void* d_ws, size_t ws_size,
                              hipStream_t stream) {
  const float* Y   = (const float*)d_in[0];  // y_hat_softmax (4,21,96,96) f32
  const float* RGB = (const float*)d_in[1];  // rgb (4,3,96,96) f32
  float* out = (float*)d_out;                // scalar loss

  zero_out_kernel<<<1, 1, 0, stream>>>(out);
  const int blocks = kN * kH * (kW / 32);    // 4*96*3 = 1152 blocks, 2 waves each
  crf_loss_wmma<<<blocks, 64, 0, stream>>>(Y, RGB, out);
}